// Sparse4DHead1st_88691074662743
// MI455X (gfx1250) — compile-verified
//
#include <hip/hip_runtime.h>
#include <cstdint>

// Problem constants (fixed by the reference harness)
constexpr int CAMS  = 6;
constexpr int C     = 256;
constexpr int NA    = 900;
constexpr int P     = 13;
constexpr int L     = 4;
constexpr int G     = 8;
constexpr int PC    = P * CAMS;    // 78 (pt,cam) pairs
constexpr int NE    = PC * L;      // 312 (pt,cam,level) entries
constexpr int WPA   = NE * G;      // 2496 weights per anchor
constexpr int HWTOT = 14960;       // sum of H*W over levels

typedef unsigned int v4u __attribute__((ext_vector_type(4)));
typedef int          v8i __attribute__((ext_vector_type(8)));
typedef int          v4i __attribute__((ext_vector_type(4)));

__global__ __launch_bounds__(256) void daf_kernel(
    const float* __restrict__ feature,   // [1, 6, 14960, 256]
    const int*   __restrict__ shapes,    // [4, 2] (H, W)
    const int*   __restrict__ starts,    // [4]
    const float* __restrict__ points,    // [1, 900, 13, 6, 2]
    const float* __restrict__ weights,   // [1, 900, 13, 6, 4, 8]
    float*       __restrict__ out)       // [1, 900, 256]
{
    __shared__ float  wlds[WPA];     // 9984 B : per-anchor attention weights
    __shared__ uint2  meta[NE * 4];  // 9984 B : {feature row, bilinear cw} per corner
    __shared__ float  pts[PC * 2];   //  624 B : per-anchor sample points
    __shared__ float4 red[256];      // 4096 B : cross-corner reduction

    const int t = threadIdx.x;
    const int a = blockIdx.x;

    // ---- Phase 0a: async-copy this anchor's points into LDS (ASYNCcnt path) ----
    // 78 * 2 floats = 624 B = 39 x b128 chunks, one per lane.
    if (t < 39) {
        unsigned ldsa = (unsigned)(uintptr_t)(&pts[0]) + (unsigned)t * 16u;
        unsigned voff = (unsigned)t * 16u;
        const float* gp = points + (size_t)a * (PC * 2);
        asm volatile("global_load_async_to_lds_b128 %0, %1, %2"
                     :: "v"(ldsa), "v"(voff), "s"(gp) : "memory");
    }

    // ---- Phase 0b: TDM DMA of this anchor's 2496 weights into LDS ----
    // D#: 2D tensor, data_size=4B, one row of tile_dim0=2496 elements.
#if __has_builtin(__builtin_amdgcn_tensor_load_to_lds)
    if (t < 32) {   // wave 0 only issues the DMA (EXEC ignored, per-wave op)
        const uint64_t ga   = (uint64_t)(uintptr_t)(weights + (size_t)a * WPA);
        const unsigned ldsw = (unsigned)(uintptr_t)(&wlds[0]);
        // group0: count=1 | lds_addr | global_addr[56:0] | type=2 (bits 127:126)
        v4u g0 = { 1u, ldsw, (unsigned)ga,
                   ((unsigned)(ga >> 32) & 0x01FFFFFFu) | 0x80000000u };
        // group1: data_size=2(4B) @ [17:16]; tensor_dim0=2496 @ [79:48];
        //         tensor_dim1=1 @ [111:80]; tile_dim0=2496 @ [127:112];
        //         tile_dim1=1 @ [143:128]; tensor_dim0_stride=2496 @ [207:160]
        v8i g1 = { 0x00020000,
                   (int)((unsigned)WPA << 16),   // tensor_dim0[15:0] << 16
                   (int)(1u << 16),              // tensor_dim1[15:0]
                   (int)((unsigned)WPA << 16),   // tile_dim0
                   1,                            // tile_dim1
                   WPA,                          // tensor_dim0_stride[31:0]
                   0, 0 };
        v4i z4 = {0, 0, 0, 0};
        v8i z8 = {0, 0, 0, 0, 0, 0, 0, 0};
        __builtin_amdgcn_tensor_load_to_lds(g0, g1, z4, z4, z8, 0);
    }
#else
    // fallback: cooperative vector copy
    {
        const float4* gw = (const float4*)(weights + (size_t)a * WPA);
        float4* lw = (float4*)wlds;
        for (int i = t; i < WPA / 4; i += 256) lw[i] = gw[i];
    }
#endif

    asm volatile("s_wait_asynccnt 0" ::: "memory");
    __syncthreads();   // points visible to all waves

    // ---- Phase 1: bilinear metadata (overlaps the TDM weight DMA) ----
    for (int e = t; e < NE; e += 256) {
        const int pc  = e >> 2;       // (p,c) pair: pc = p*6 + c
        const int l   = e & 3;        // level
        const int cam = pc % CAMS;
        const float px = pts[pc * 2 + 0];
        const float py = pts[pc * 2 + 1];
        const int h  = shapes[l * 2 + 0];
        const int w  = shapes[l * 2 + 1];
        const int s0 = starts[l];
        const float x  = px * (float)w - 0.5f;
        const float y  = py * (float)h - 0.5f;
        const float xf = floorf(x), yf = floorf(y);
        const int   x0 = (int)xf,   y0 = (int)yf;
        const float fx = x - xf,    fy = y - yf;
        const float wx[2] = {1.0f - fx, fx};
        const float wy[2] = {1.0f - fy, fy};
        const int base = cam * HWTOT + s0;
        #pragma unroll
        for (int corner = 0; corner < 4; ++corner) {
            const int dx = corner & 1, dy = corner >> 1;
            const int xi = x0 + dx,   yi = y0 + dy;
            const bool valid = (xi >= 0) & (xi < w) & (yi >= 0) & (yi < h);
            const int xc = min(max(xi, 0), w - 1);
            const int yc = min(max(yi, 0), h - 1);
            const float cw = valid ? wx[dx] * wy[dy] : 0.0f;
            meta[e * 4 + corner] =
                uint2{(unsigned)(base + yc * w + xc), __float_as_uint(cw)};
        }
    }

#if __has_builtin(__builtin_amdgcn_s_wait_tensorcnt)
    __builtin_amdgcn_s_wait_tensorcnt(0);
#endif
    __syncthreads();   // meta + weights visible

    // ---- Phase 2: coalesced gather + grouped weighted accumulation ----
    const int corner = t >> 6;          // wave-uniform -> LDS broadcast reads
    const int ch     = (t & 63) * 4;    // 4 channels per lane
    const int g      = ch >> 5;         // channel group
    const float* __restrict__ fbase = feature + ch;
    float4 acc = {0.f, 0.f, 0.f, 0.f};
    #pragma unroll 4
    for (int e = 0; e < NE; ++e) {
        const uint2 m  = meta[(e << 2) + corner];       // ds_load_b64 broadcast
        const float s  = __uint_as_float(m.y) * wlds[e * 8 + g];
        const float4 v = *(const float4*)(fbase + (size_t)m.x * C);  // b128, coalesced
        acc.x = fmaf(s, v.x, acc.x);
        acc.y = fmaf(s, v.y, acc.y);
        acc.z = fmaf(s, v.z, acc.z);
        acc.w = fmaf(s, v.w, acc.w);
    }

    // ---- Phase 3: reduce the 4 corner partials, store out[a, :] ----
    red[t] = acc;
    __syncthreads();
    if (t < 64) {
        const float4 r0 = red[t], r1 = red[t + 64], r2 = red[t + 128], r3 = red[t + 192];
        float4 r;
        r.x = (r0.x + r1.x) + (r2.x + r3.x);
        r.y = (r0.y + r1.y) + (r2.y + r3.y);
        r.z = (r0.z + r1.z) + (r2.z + r3.z);
        r.w = (r0.w + r1.w) + (r2.w + r3.w);
        *(float4*)(out + (size_t)a * C + t * 4) = r;
    }
}

extern "C" void kernel_launch(void* const* d_in, const int* in_sizes, int n_in,
                              void* d_out, int out_size, void* d_ws, size_t ws_size,
                              hipStream_t stream) {
    const float* feature = (const float*)d_in[0];
    const int*   shapes  = (const int*)d_in[1];
    const int*   starts  = (const int*)d_in[2];
    const float* points  = (const float*)d_in[3];
    const float* weights = (const float*)d_in[4];
    float* out = (float*)d_out;
    daf_kernel<<<NA, 256, 0, stream>>>(feature, shapes, starts, points, weights, out);
}